// INLayer_17970143166937
// MI455X (gfx1250) — compile-verified
//
#include <hip/hip_runtime.h>

typedef __attribute__((ext_vector_type(2))) float v2f;
typedef __attribute__((ext_vector_type(8))) float v8f;

#define NN 50000
#define NE 1600000
#define DIM 16
#define HID 64

__device__ __forceinline__ v8f wmma_f32(v2f a, v2f b, v8f c) {
  // D = A(16x4) * B(4x16) + C(16x16), fp32, wave32
  return __builtin_amdgcn_wmma_f32_16x16x4_f32(false, a, false, b, (short)0, c,
                                               false, false);
}

__global__ void zero_f32(float* __restrict__ p, int n) {
  int i = blockIdx.x * blockDim.x + threadIdx.x;
  if (i < n) p[i] = 0.0f;
}

// Per-edge MLP: e_new = ReLU([x[dst]|x[src]|e] @ W1 + b1) @ W2 + b2
// + atomic scatter-sum of e_new into agg[dst].
__global__ __launch_bounds__(256) void edge_mlp_scatter(
    const float* __restrict__ x, const long long* __restrict__ eidx,
    const float* __restrict__ e, const float* __restrict__ W1,
    const float* __restrict__ b1, const float* __restrict__ W2,
    const float* __restrict__ b2, float* __restrict__ e_out,
    float* __restrict__ agg) {
  __shared__ float lds[8 * 16 * 68];  // per-wave 16x64 H tile, stride 68 (bank-conflict-free)
  const int lane = threadIdx.x & 31;
  const int wv = threadIdx.x >> 5;
  const int lo = lane & 15;  // row (A/M) or col (B,N / D,N) within tile
  const int hi = lane >> 4;  // K-half select
  float* hbuf = &lds[wv * 16 * 68];

  // B-layout weight fragments, resident in VGPRs across the whole tile loop.
  // B frag element: vgpr v, lane L -> W[4k + v + 2*hi][col]
  v2f w1f[12][4];
#pragma unroll
  for (int k = 0; k < 12; ++k)
#pragma unroll
    for (int n = 0; n < 4; ++n) {
      w1f[k][n][0] = W1[(4 * k + 2 * hi + 0) * HID + 16 * n + lo];
      w1f[k][n][1] = W1[(4 * k + 2 * hi + 1) * HID + 16 * n + lo];
    }
  v2f w2f[16];
#pragma unroll
  for (int k = 0; k < 16; ++k) {
    w2f[k][0] = W2[(4 * k + 2 * hi + 0) * DIM + lo];
    w2f[k][1] = W2[(4 * k + 2 * hi + 1) * DIM + lo];
  }
  float b1v[4];
#pragma unroll
  for (int n = 0; n < 4; ++n) b1v[n] = b1[16 * n + lo];
  const float b2v = b2[lo];

  const int tiles = NE / 16;
  const int wid = blockIdx.x * 8 + wv;
  const int wstep = gridDim.x * 8;
  for (int t = wid; t < tiles; t += wstep) {
    const int base = t * 16;
    const int me = base + lo;  // this lane's edge (rows of the A tile)
    const int srcIdx = (int)eidx[me];
    const int dstIdx = (int)eidx[NE + me];
    const float* xdst = x + dstIdx * DIM + 2 * hi;
    const float* xsrc = x + srcIdx * DIM + 2 * hi;
    const float* erow = e + me * DIM + 2 * hi;

    // H[16 edges x 64 hidden] accumulators, bias-initialized
    v8f acc[4];
#pragma unroll
    for (int n = 0; n < 4; ++n)
#pragma unroll
      for (int i = 0; i < 8; ++i) acc[n][i] = b1v[n];

    // GEMM1: 12 K-steps (feature concat: x[dst] | x[src] | e)
#pragma unroll
    for (int k = 0; k < 12; ++k) {
      const float* ap = (k < 4)   ? (xdst + 4 * k)
                        : (k < 8) ? (xsrc + 4 * (k - 4))
                                  : (erow + 4 * (k - 8));
      v2f a = *(const v2f*)ap;  // global_load_b64 gather, A-layout
#pragma unroll
      for (int n = 0; n < 4; ++n) acc[n] = wmma_f32(a, w1f[k][n], acc[n]);
    }

    // ReLU + stage H to LDS (D-layout -> row-major tile, stride 68)
#pragma unroll
    for (int n = 0; n < 4; ++n)
#pragma unroll
      for (int w = 0; w < 8; ++w) {
        float h = acc[n][w];
        h = h > 0.0f ? h : 0.0f;
        hbuf[(w + 8 * hi) * 68 + 16 * n + lo] = h;
      }

    // GEMM2: e_tile[16x16] = H @ W2 + b2, A-frags re-read from LDS (transposed view)
    v8f acc2;
#pragma unroll
    for (int i = 0; i < 8; ++i) acc2[i] = b2v;
#pragma unroll
    for (int k = 0; k < 16; ++k) {
      v2f a2 = *(const v2f*)&hbuf[lo * 68 + 4 * k + 2 * hi];  // ds_load_b64
      acc2 = wmma_f32(a2, w2f[k], acc2);
    }

    // Store e_new (coalesced) + atomic scatter into agg[dst]
#pragma unroll
    for (int w = 0; w < 8; ++w) {
      const int row = base + w + 8 * hi;
      e_out[row * DIM + lo] = acc2[w];
      const int d = __shfl(dstIdx, w + 8 * hi, 32);
      atomicAdd(&agg[d * DIM + lo], acc2[w]);
    }
  }
}

// Per-node MLP: x_new = ReLU([x|agg] @ W1 + b1) @ W2 + b2
__global__ __launch_bounds__(256) void node_mlp(
    const float* __restrict__ x, const float* __restrict__ agg,
    const float* __restrict__ W1, const float* __restrict__ b1,
    const float* __restrict__ W2, const float* __restrict__ b2,
    float* __restrict__ x_out) {
  __shared__ float lds[8 * 16 * 68];
  const int lane = threadIdx.x & 31;
  const int wv = threadIdx.x >> 5;
  const int lo = lane & 15;
  const int hi = lane >> 4;
  float* hbuf = &lds[wv * 16 * 68];

  v2f w1f[8][4];
#pragma unroll
  for (int k = 0; k < 8; ++k)
#pragma unroll
    for (int n = 0; n < 4; ++n) {
      w1f[k][n][0] = W1[(4 * k + 2 * hi + 0) * HID + 16 * n + lo];
      w1f[k][n][1] = W1[(4 * k + 2 * hi + 1) * HID + 16 * n + lo];
    }
  v2f w2f[16];
#pragma unroll
  for (int k = 0; k < 16; ++k) {
    w2f[k][0] = W2[(4 * k + 2 * hi + 0) * DIM + lo];
    w2f[k][1] = W2[(4 * k + 2 * hi + 1) * DIM + lo];
  }
  float b1v[4];
#pragma unroll
  for (int n = 0; n < 4; ++n) b1v[n] = b1[16 * n + lo];
  const float b2v = b2[lo];

  const int tiles = NN / 16;
  const int wid = blockIdx.x * 8 + wv;
  const int wstep = gridDim.x * 8;
  for (int t = wid; t < tiles; t += wstep) {
    const int base = t * 16;
    const float* xrow = x + (base + lo) * DIM + 2 * hi;
    const float* arow = agg + (base + lo) * DIM + 2 * hi;

    v8f acc[4];
#pragma unroll
    for (int n = 0; n < 4; ++n)
#pragma unroll
      for (int i = 0; i < 8; ++i) acc[n][i] = b1v[n];

#pragma unroll
    for (int k = 0; k < 8; ++k) {
      const float* ap = (k < 4) ? (xrow + 4 * k) : (arow + 4 * (k - 4));
      v2f a = *(const v2f*)ap;
#pragma unroll
      for (int n = 0; n < 4; ++n) acc[n] = wmma_f32(a, w1f[k][n], acc[n]);
    }

#pragma unroll
    for (int n = 0; n < 4; ++n)
#pragma unroll
      for (int w = 0; w < 8; ++w) {
        float h = acc[n][w];
        h = h > 0.0f ? h : 0.0f;
        hbuf[(w + 8 * hi) * 68 + 16 * n + lo] = h;
      }

    v8f acc2;
#pragma unroll
    for (int i = 0; i < 8; ++i) acc2[i] = b2v;
#pragma unroll
    for (int k = 0; k < 16; ++k) {
      v2f a2 = *(const v2f*)&hbuf[lo * 68 + 4 * k + 2 * hi];
      acc2 = wmma_f32(a2, w2f[k], acc2);
    }

#pragma unroll
    for (int w = 0; w < 8; ++w) {
      x_out[(base + w + 8 * hi) * DIM + lo] = acc2[w];
    }
  }
}

extern "C" void kernel_launch(void* const* d_in, const int* in_sizes, int n_in,
                              void* d_out, int out_size, void* d_ws,
                              size_t ws_size, hipStream_t stream) {
  const float* x = (const float*)d_in[0];
  const long long* eidx = (const long long*)d_in[1];  // int64 [2, NE]
  const float* e = (const float*)d_in[2];
  const float* fR_W1 = (const float*)d_in[3];
  const float* fR_b1 = (const float*)d_in[4];
  const float* fR_W2 = (const float*)d_in[5];
  const float* fR_b2 = (const float*)d_in[6];
  const float* fO_W1 = (const float*)d_in[7];
  const float* fO_b1 = (const float*)d_in[8];
  const float* fO_W2 = (const float*)d_in[9];
  const float* fO_b2 = (const float*)d_in[10];

  float* out = (float*)d_out;
  float* x_new = out;                 // [NN, 16]
  float* e_new = out + NN * DIM;      // [NE, 16]
  float* agg = (float*)d_ws;          // [NN, 16] scratch, zeroed every call

  const int aggN = NN * DIM;
  zero_f32<<<(aggN + 255) / 256, 256, 0, stream>>>(agg, aggN);
  edge_mlp_scatter<<<1568, 256, 0, stream>>>(x, eidx, e, fR_W1, fR_b1, fR_W2,
                                             fR_b2, e_new, agg);
  node_mlp<<<391, 256, 0, stream>>>(x, agg, fO_W1, fO_b1, fO_W2, fO_b2, x_new);
}